// TransE_SourceMatrix_37890201486007
// MI455X (gfx1250) — compile-verified
//
#include <hip/hip_runtime.h>
#include <hip/hip_bf16.h>
#include <math.h>

typedef float v2f __attribute__((ext_vector_type(2)));
typedef float v8f __attribute__((ext_vector_type(8)));

#define DIM   100
#define KSTEP 4
#define EPS   1e-12f

// ---------------------------------------------------------------- init
__global__ void k_init(int* counts, int* perm, int permLen) {
    int idx = blockIdx.x * blockDim.x + threadIdx.x;
    if (idx < 48) counts[idx] = 0;            // counts[16], cursor[16], offs[16]
    if (idx < permLen) perm[idx] = -1;
}

// ---------------------------------------------------------------- count
__global__ void k_count(const int* __restrict__ src, const int* __restrict__ srcb,
                        int* counts, int n) {
    int idx = blockIdx.x * blockDim.x + threadIdx.x;
    if (idx >= 2 * n) return;
    int s = (idx < n) ? src[idx] : srcb[idx - n];   // s in [2,16]
    atomicAdd(&counts[s - 2], 1);
}

// ---------------------------------------------------------------- padded offsets
__global__ void k_offsets(int* counts) {
    int* cursor = counts + 16;
    int* offs   = counts + 32;
    if (threadIdx.x == 0 && blockIdx.x == 0) {
        int acc = 0;
        for (int e = 0; e < 15; ++e) {
            offs[e]   = acc;
            cursor[e] = acc;
            acc += ((counts[e] + 15) / 16) * 16;     // pad each segment to 16
        }
        offs[15] = acc;                              // padded total
    }
}

// ---------------------------------------------------------------- scatter
__global__ void k_scatter(const int* __restrict__ src, const int* __restrict__ srcb,
                          int* cursor, int* perm, int n) {
    int idx = blockIdx.x * blockDim.x + threadIdx.x;
    if (idx >= 2 * n) return;
    int s = (idx < n) ? src[idx] : srcb[idx - n];
    int pos = atomicAdd(&cursor[s - 2], 1);
    perm[pos] = idx;
}

// ---------------------------------------------------------------- main GEMM + score
// One block = one tile of 16 triples (single expert). 8 waves (256 thr).
// Waves 0..6: output-dim tiles n0 = 16*wave (dims 100..111 are junk columns,
// computed on clamped in-bounds weights and simply never stored).
__global__ void __launch_bounds__(256)
k_transe(const int* __restrict__ heads,  const int* __restrict__ rels,
         const int* __restrict__ tails,
         const int* __restrict__ heads_b, const int* __restrict__ rels_b,
         const int* __restrict__ tails_b,
         const float* __restrict__ ents_emb, const float* __restrict__ rels_emb,
         const float* __restrict__ entW, const float* __restrict__ entB,
         const float* __restrict__ relW, const float* __restrict__ relB,
         const int* __restrict__ perm, const int* __restrict__ offs,
         float* __restrict__ out, int n) {

    __shared__ float sA[3][16][104];   // staged embeddings: 0=h, 1=r, 2=t
    __shared__ float sO[3][16][DIM];   // tanh outputs:      0=hv,1=rv,2=tv
    __shared__ int   sId[3][16];       // row ids (-1 = padded triple)

    const int tid       = threadIdx.x;
    const int tile_base = blockIdx.x * 16;
    const int total     = offs[15];
    if (tile_base >= total) return;

    // uniform expert for this tile (segments are 16-aligned)
    int e = 0;
    #pragma unroll
    for (int i = 0; i < 15; ++i)
        if (tile_base >= offs[i + 1]) e = i + 1;

    // ---------------- stage 48 row ids ---------------------------------------
    if (tid < 48) {
        int vec = tid >> 4;            // 0=h, 1=r, 2=t
        int m   = tid & 15;
        int gi  = perm[tile_base + m];
        int id  = -1;
        if (gi >= 0) {
            if (vec == 0)      id = (gi < n) ? heads[gi] : heads_b[gi - n];
            else if (vec == 1) id = (gi < n) ? rels[gi]  : rels_b[gi - n];
            else               id = (gi < n) ? tails[gi] : tails_b[gi - n];
        }
        sId[vec][m] = id;
    }
    __syncthreads();

    // ---------------- gather 3 x 16 rows as float4 (b128) --------------------
    // 400B rows are 16B aligned (id*400); 25 float4 per row; 1200 total.
    for (int idx = tid; idx < 3 * 16 * 25; idx += 256) {
        int vec = idx / 400;
        int rem = idx - vec * 400;
        int m   = rem / 25;
        int c4  = rem - m * 25;
        int id  = sId[vec][m];
        float4 v = make_float4(0.f, 0.f, 0.f, 0.f);
        if (id >= 0) {
            const float* table = (vec == 1) ? rels_emb : ents_emb;
            v = *((const float4*)(table + (size_t)id * DIM) + c4);
        }
        *(float4*)&sA[vec][m][c4 * 4] = v;
    }
    __syncthreads();

    const int wave = tid >> 5;
    const int lane = tid & 31;

    if (wave < 7) {
        const int n0    = wave * 16;
        const int nl    = lane & 15;          // N index within tile / A row
        const int khalf = (lane >> 4) * 2;    // lanes 16-31 carry K+2,K+3
        const int ncol  = n0 + nl;            // global output dim
        const bool nok  = (ncol < DIM);
        const int  nc   = nok ? ncol : (DIM - 1);   // in-bounds clamp: junk
                                                    // columns never stored

        // weight rows for this lane's output column (branch-free, EXEC stays 1s)
        const float* We = entW + (size_t)e * DIM * DIM + (size_t)nc * DIM;
        const float* Wr = relW + (size_t)e * DIM * DIM + (size_t)nc * DIM;

        v8f ah = {0.f,0.f,0.f,0.f,0.f,0.f,0.f,0.f};
        v8f ar = ah, at = ah;

        #pragma unroll
        for (int k0 = 0; k0 < DIM; k0 += KSTEP) {
            const int k = k0 + khalf;
            v2f aH, aR, aT, bE, bR;
            aH[0] = sA[0][nl][k];  aH[1] = sA[0][nl][k + 1];
            aR[0] = sA[1][nl][k];  aR[1] = sA[1][nl][k + 1];
            aT[0] = sA[2][nl][k];  aT[1] = sA[2][nl][k + 1];
            bE[0] = We[k];         bE[1] = We[k + 1];
            bR[0] = Wr[k];         bR[1] = Wr[k + 1];
            // D = A x B + C ; hv and tv reuse the ent-weight B fragment
            ah = __builtin_amdgcn_wmma_f32_16x16x4_f32(false, aH, false, bE,
                                                       (short)0, ah, false, false);
            at = __builtin_amdgcn_wmma_f32_16x16x4_f32(false, aT, false, bE,
                                                       (short)0, at, false, false);
            ar = __builtin_amdgcn_wmma_f32_16x16x4_f32(false, aR, false, bR,
                                                       (short)0, ar, false, false);
        }

        if (nok) {
            const float be = entB[(size_t)e * DIM + ncol];
            const float br = relB[(size_t)e * DIM + ncol];
            const int mbase = (lane >> 4) * 8;   // C/D: VGPR i -> rows i, i+8
            #pragma unroll
            for (int i = 0; i < 8; ++i) {
                int m = i + mbase;
                sO[0][m][ncol] = tanhf(ah[i] + be);
                sO[1][m][ncol] = tanhf(ar[i] + br);
                sO[2][m][ncol] = tanhf(at[i] + be);
            }
        }
    }
    __syncthreads();

    // ---------------- phase 2: normalize + score ----------------------------
    // 16 threads per triple (within a half-wave), strided over the 100 dims,
    // reduced with wave32 xor-butterflies (width 16).
    {
        const int m = tid >> 4;        // triple 0..15
        const int j = tid & 15;        // dim-group lane

        float sh = 0.f, sr = 0.f, st = 0.f;
        for (int c = j; c < DIM; c += 16) {
            float h = sO[0][m][c], r = sO[1][m][c], t = sO[2][m][c];
            sh += h * h; sr += r * r; st += t * t;
        }
        #pragma unroll
        for (int d = 1; d < 16; d <<= 1) {
            sh += __shfl_xor(sh, d, 16);
            sr += __shfl_xor(sr, d, 16);
            st += __shfl_xor(st, d, 16);
        }
        float ih = 1.0f / fmaxf(sqrtf(sh), EPS);
        float ir = 1.0f / fmaxf(sqrtf(sr), EPS);
        float it = 1.0f / fmaxf(sqrtf(st), EPS);

        float ss = 0.f;
        for (int c = j; c < DIM; c += 16) {
            float d = sO[0][m][c] * ih + sO[1][m][c] * ir - sO[2][m][c] * it;
            ss += d * d;
        }
        #pragma unroll
        for (int d = 1; d < 16; d <<= 1)
            ss += __shfl_xor(ss, d, 16);

        if (j == 0) {
            int gi = perm[tile_base + m];
            if (gi >= 0)
                out[gi] = sqrtf(ss);   // concat order == (scores0, scores1) flat
        }
    }
}

// ---------------------------------------------------------------- launcher
extern "C" void kernel_launch(void* const* d_in, const int* in_sizes, int n_in,
                              void* d_out, int out_size, void* d_ws, size_t ws_size,
                              hipStream_t stream) {
    const int*   heads    = (const int*)  d_in[0];
    const int*   rels     = (const int*)  d_in[1];
    const int*   tails    = (const int*)  d_in[2];
    const int*   sources  = (const int*)  d_in[3];
    const int*   heads_b  = (const int*)  d_in[4];
    const int*   rels_b   = (const int*)  d_in[5];
    const int*   tails_b  = (const int*)  d_in[6];
    const int*   sources_b= (const int*)  d_in[7];
    const float* ents_emb = (const float*)d_in[8];
    const float* rels_emb = (const float*)d_in[9];
    const float* entW     = (const float*)d_in[10];
    const float* entB     = (const float*)d_in[11];
    const float* relW     = (const float*)d_in[12];
    const float* relB     = (const float*)d_in[13];
    float*       out      = (float*)d_out;

    const int n   = in_sizes[0];          // N = 65536
    const int two = 2 * n;

    int* counts = (int*)d_ws;             // [16]
    int* cursor = counts + 16;            // [16]
    int* offs   = counts + 32;            // [16]  offs[15] = padded total
    int* perm   = counts + 64;            // [2N + 256]
    const int permLen = two + 256;

    const int maxTiles = (two + 15 * 15 + 15) / 16;   // 8207 for N=65536

    k_init   <<<(permLen + 255) / 256, 256, 0, stream>>>(counts, perm, permLen);
    k_count  <<<(two + 255) / 256,     256, 0, stream>>>(sources, sources_b, counts, n);
    k_offsets<<<1, 32, 0, stream>>>(counts);
    k_scatter<<<(two + 255) / 256,     256, 0, stream>>>(sources, sources_b, cursor, perm, n);
    k_transe <<<maxTiles, 256, 0, stream>>>(heads, rels, tails,
                                            heads_b, rels_b, tails_b,
                                            ents_emb, rels_emb,
                                            entW, entB, relW, relB,
                                            perm, offs, out, n);
}